// l1_ssim_loss_27479200759828
// MI455X (gfx1250) — compile-verified
//
#include <hip/hip_runtime.h>
#include <cmath>

typedef float v2f __attribute__((ext_vector_type(2)));
typedef float v8f __attribute__((ext_vector_type(8)));

#define IMG_W   512
#define VALID   502   // 512 - 11 + 1

struct W11 { float w[11]; };
struct W13 { float w[13]; };

static __device__ __forceinline__ v8f wmma_k4(v2f a, v2f b, v8f c) {
    // D(16x16,f32) = A(16x4,f32) * B(4x16,f32) + C
    return __builtin_amdgcn_wmma_f32_16x16x4_f32(false, a, false, b, (short)0, c, false, false);
}

// ---------------------------------------------------------------------------
// Kernel 0: zero the f64 SSIM accumulator in workspace
// ---------------------------------------------------------------------------
__global__ void zero_acc_kernel(double* acc) {
    if (threadIdx.x == 0) acc[0] = 0.0;
}

// ---------------------------------------------------------------------------
// Kernel 1: SSIM tile kernel. One wave32 per 16x16 output tile.
// Valid (unpadded) separable 11-tap gaussian conv of {x,y,xx,yy,xy} via WMMA,
// SSIM formula, masked sum -> f64 atomicAdd.
// ---------------------------------------------------------------------------
__global__ __launch_bounds__(32) void ssim_kernel(
    const float* __restrict__ x, const float* __restrict__ y,
    double* __restrict__ acc, W11 wk)
{
    __shared__ float sx[32][28];        // input tile x (rows 26..31, cols 26..27 zero)
    __shared__ float sy[32][28];        // input tile y
    __shared__ float sWt[16][34];       // band weights, transposed: sWt[n][k] = w[k-n]
    __shared__ float sH5[5][16][34];    // horizontal results, transposed: [map][n][k]
    __shared__ float sred[32];

    const int lane = threadIdx.x;       // 0..31, full wave
    const int ln   = lane & 15;         // row index of A / col index of B,D
    const int half = lane >> 4;
    const int tc0  = blockIdx.x * 16;
    const int tr0  = blockIdx.y * 16;
    const size_t ib = (size_t)blockIdx.z * (IMG_W * IMG_W);
    const float* xb = x + ib;
    const float* yb = y + ib;

    // zero-fill input tiles (avoid NaN garbage feeding WMMA through zero weights)
    {
        float* px = &sx[0][0];
        float* py = &sy[0][0];
        for (int i = lane; i < 32 * 28; i += 32) { px[i] = 0.0f; py[i] = 0.0f; }
    }
    // band weight matrix (transposed): sWt[n][k] = w[k-n] for 0<=k-n<=10
    for (int i = lane; i < 16 * 28; i += 32) {
        int n = i / 28, k = i % 28, d = k - n;
        sWt[n][k] = (d >= 0 && d <= 10) ? wk.w[d] : 0.0f;
    }
    __syncthreads();
    // load 26x26 tile, clamped at image edge (clamped values only reach masked outputs)
    for (int i = lane; i < 26 * 26; i += 32) {
        int r = i / 26, c = i % 26;
        int gr = tr0 + r; if (gr > IMG_W - 1) gr = IMG_W - 1;
        int gc = tc0 + c; if (gc > IMG_W - 1) gc = IMG_W - 1;
        size_t idx = (size_t)gr * IMG_W + gc;
        sx[r][c] = xb[idx];
        sy[r][c] = yb[idx];
    }
    __syncthreads();

    // ---- horizontal pass: Hout[k=0..31][n] for all 5 maps ----
    for (int mb = 0; mb < 2; ++mb) {
        const int rb = mb * 16;
        v8f a0 = {}, a1 = {}, a2 = {}, a3 = {}, a4 = {};
        for (int kc = 0; kc < 7; ++kc) {
            const int kb = kc * 4 + 2 * half;           // this lane's K pair
            v2f ax = *(const v2f*)&sx[rb + ln][kb];     // A[m][k], m=ln
            v2f ay = *(const v2f*)&sy[rb + ln][kb];
            v2f b  = *(const v2f*)&sWt[ln][kb];         // B[k][n]=w[k-n], n=ln
            a0 = wmma_k4(ax,      b, a0);               // mean(x)
            a1 = wmma_k4(ay,      b, a1);               // mean(y)
            a2 = wmma_k4(ax * ax, b, a2);               // mean(xx)
            a3 = wmma_k4(ay * ay, b, a3);               // mean(yy)
            a4 = wmma_k4(ax * ay, b, a4);               // mean(xy)
        }
        // D layout: vgpr v, lane -> (row = v + 8*half, col = ln); store transposed
        for (int v = 0; v < 8; ++v) {
            int row = rb + v + 8 * half;
            sH5[0][ln][row] = a0[v];
            sH5[1][ln][row] = a1[v];
            sH5[2][ln][row] = a2[v];
            sH5[3][ln][row] = a3[v];
            sH5[4][ln][row] = a4[v];
        }
    }
    __syncthreads();

    // ---- vertical pass: Out = Wv(16x28) * Hout(28x16); Wv[m][k]=w[k-m]=sWt[m][k] ----
    v8f res[5];
    for (int map = 0; map < 5; ++map) {
        v8f c8 = {};
        for (int kc = 0; kc < 7; ++kc) {
            const int kb = kc * 4 + 2 * half;
            v2f a = *(const v2f*)&sWt[ln][kb];          // A[m][k], m=ln
            v2f b = *(const v2f*)&sH5[map][ln][kb];     // B[k][n], n=ln
            c8 = wmma_k4(a, b, c8);
        }
        res[map] = c8;
    }

    // ---- SSIM formula + masked tile sum ----
    const float c1 = 1.0e-4f, c2 = 9.0e-4f;
    float lsum = 0.0f;
    for (int v = 0; v < 8; ++v) {
        float mx  = res[0][v], my  = res[1][v];
        float mxx = res[2][v], myy = res[3][v], mxy = res[4][v];
        float vx  = mxx - mx * mx;
        float vy  = myy - my * my;
        float cov = mxy - mx * my;
        float s = ((2.0f * mx * my + c1) * (2.0f * cov + c2)) /
                  ((mx * mx + my * my + c1) * (vx + vy + c2));
        int orow = tr0 + v + 8 * half;
        int ocol = tc0 + ln;
        if (orow < VALID && ocol < VALID) lsum += s;
    }
    sred[lane] = lsum;
    __syncthreads();
    if (lane == 0) {
        float t = 0.0f;
        for (int i = 0; i < 32; ++i) t += sred[i];
        atomicAdd(acc, (double)t);
    }
}

// ---------------------------------------------------------------------------
// Kernel 2: L1 branch + combine. filt(y)-filt(x) == filt(y-x) (linearity).
// SAME zero-padded separable 13-tap erf-gaussian conv via WMMA.
// ---------------------------------------------------------------------------
__global__ __launch_bounds__(32) void l1_kernel(
    const float* __restrict__ x, const float* __restrict__ y,
    float* __restrict__ out, const double* __restrict__ acc, W13 wk)
{
    __shared__ float sd[32][28];        // (y-x) tile, zero-padded borders
    __shared__ float sWt[16][34];       // sWt[n][k] = w[k-n]
    __shared__ float sHt[16][34];       // horizontal result, transposed

    const int lane = threadIdx.x;
    const int ln   = lane & 15;
    const int half = lane >> 4;
    const int tc0  = blockIdx.x * 16;
    const int tr0  = blockIdx.y * 16;
    const size_t ib = (size_t)blockIdx.z * (IMG_W * IMG_W);
    const float* xb = x + ib;
    const float* yb = y + ib;

    {
        float* pd = &sd[0][0];
        for (int i = lane; i < 32 * 28; i += 32) pd[i] = 0.0f;
    }
    for (int i = lane; i < 16 * 28; i += 32) {
        int n = i / 28, k = i % 28, d = k - n;
        sWt[n][k] = (d >= 0 && d <= 12) ? wk.w[d] : 0.0f;
    }
    __syncthreads();
    // 28x28 input tile, SAME padding (pad=6) with zeros outside the image
    for (int i = lane; i < 28 * 28; i += 32) {
        int r = i / 28, c = i % 28;
        int gr = tr0 + r - 6, gc = tc0 + c - 6;
        float v = 0.0f;
        if (gr >= 0 && gr < IMG_W && gc >= 0 && gc < IMG_W) {
            size_t idx = (size_t)gr * IMG_W + gc;
            v = yb[idx] - xb[idx];
        }
        sd[r][c] = v;
    }
    __syncthreads();

    // horizontal pass
    for (int mb = 0; mb < 2; ++mb) {
        const int rb = mb * 16;
        v8f c8 = {};
        for (int kc = 0; kc < 7; ++kc) {
            const int kb = kc * 4 + 2 * half;
            v2f a = *(const v2f*)&sd[rb + ln][kb];
            v2f b = *(const v2f*)&sWt[ln][kb];
            c8 = wmma_k4(a, b, c8);
        }
        for (int v = 0; v < 8; ++v) sHt[ln][rb + v + 8 * half] = c8[v];
    }
    __syncthreads();

    // vertical pass
    v8f c8 = {};
    for (int kc = 0; kc < 7; ++kc) {
        const int kb = kc * 4 + 2 * half;
        v2f a = *(const v2f*)&sWt[ln][kb];
        v2f b = *(const v2f*)&sHt[ln][kb];
        c8 = wmma_k4(a, b, c8);
    }

    // combine with SSIM scalar
    const float ssim_mean = (float)(acc[0] / (64.0 * (double)VALID * (double)VALID));
    const float ssim_loss = 1.0f - ssim_mean;
    const float base  = 100.0f * 0.84f * ssim_loss;
    const float scale = 100.0f * 0.16f;
    float* ob = out + ib;
    for (int v = 0; v < 8; ++v) {
        int orow = tr0 + v + 8 * half;
        int ocol = tc0 + ln;
        ob[(size_t)orow * IMG_W + ocol] = base + scale * fabsf(c8[v]);
    }
}

// ---------------------------------------------------------------------------
extern "C" void kernel_launch(void* const* d_in, const int* in_sizes, int n_in,
                              void* d_out, int out_size, void* d_ws, size_t ws_size,
                              hipStream_t stream) {
    (void)in_sizes; (void)n_in; (void)out_size; (void)ws_size;
    const float* x = (const float*)d_in[0];
    const float* y = (const float*)d_in[1];
    float* out = (float*)d_out;
    double* acc = (double*)d_ws;

    // SSIM 11-tap normalized gaussian, sigma=1.5
    W11 w11;
    {
        double s = 0.0, g[11];
        for (int i = 0; i < 11; ++i) { double d = i - 5; g[i] = std::exp(-(d * d) / 4.5); s += g[i]; }
        for (int i = 0; i < 11; ++i) w11.w[i] = (float)(g[i] / s);
    }
    // 13-tap erf-gaussian (monai gaussian_1d, approx='erf', unnormalized), sigma=1.5
    W13 w13;
    {
        const double t = 0.70710678 / 1.5;
        for (int i = 0; i < 13; ++i) {
            double xi = i - 6;
            double v = 0.5 * (std::erf(t * (xi + 0.5)) - std::erf(t * (xi - 0.5)));
            w13.w[i] = (float)(v < 0.0 ? 0.0 : v);
        }
    }

    dim3 grid(IMG_W / 16, IMG_W / 16, 64);
    zero_acc_kernel<<<1, 32, 0, stream>>>(acc);
    ssim_kernel<<<grid, 32, 0, stream>>>(x, y, acc, w11);
    l1_kernel<<<grid, 32, 0, stream>>>(x, y, out, acc, w13);
}